// RoleContrastLoss_79671643341677
// MI455X (gfx1250) — compile-verified
//
#include <hip/hip_runtime.h>
#include <math.h>

// CDNA5 / gfx1250 wave32 WMMA types
typedef float v2f __attribute__((ext_vector_type(2)));
typedef float v8f __attribute__((ext_vector_type(8)));

#define D_DIM   1024
#define NR      4096            // 256 events * 16 roles
#define NTILES  (NR / 16)       // 256 column tiles
#define KCH     (D_DIM / 4)     // 256 K-chunks of 4 for wmma f32 16x16x4
#define LDS_STRIDE (D_DIM + 4)  // +4 floats: lanes hit banks 4*n (conflict-free b64)
#define WAVES   8
#define TOPK    5

__device__ __forceinline__ void top5_insert(float t[TOPK], float v) {
    // branch-free insertion into descending sorted 5-list
    float w;
    w = fminf(t[0], v); t[0] = fmaxf(t[0], v); v = w;
    w = fminf(t[1], v); t[1] = fmaxf(t[1], v); v = w;
    w = fminf(t[2], v); t[2] = fmaxf(t[2], v); v = w;
    w = fminf(t[3], v); t[3] = fmaxf(t[3], v); v = w;
    t[4] = fmaxf(t[4], v);
}

__global__ void __launch_bounds__(256)
role_loss_main(const float* __restrict__ X, float* __restrict__ gAcc) {
    __shared__ float As[16 * LDS_STRIDE];                  // ~65.8 KB A strip
    __shared__ float topBuf[WAVES * 16 * 16 * TOPK];       // 40 KB per-lane top5
    __shared__ float mergeBuf[16 * 16 * TOPK];             // 5 KB
    __shared__ float posL[16], negL[16];

    const int tid  = threadIdx.x;
    const int lane = tid & 31;
    const int wave = tid >> 5;
    const int n    = lane & 15;   // N-index within tile (and A row index for A-frag)
    const int h    = lane >> 4;   // lane half -> K offset +2, and C rows +8
    const int it   = blockIdx.x;  // row-tile index
    const int i0   = it * 16;

    if (tid < 16) { posL[tid] = 0.0f; negL[tid] = 0.0f; }

    // cooperative load of the 16x1024 A strip into padded LDS (float4 vectorized)
    for (int row = 0; row < 16; ++row) {
        const float4* src = (const float4*)(X + (size_t)(i0 + row) * D_DIM);
        float4* dst = (float4*)(As + row * LDS_STRIDE);
        for (int q = tid; q < D_DIM / 4; q += 256) dst[q] = src[q];
    }
    __syncthreads();

    float pos[8], neg[8], t5[8][TOPK];
#pragma unroll
    for (int r = 0; r < 8; ++r) {
        pos[r] = 0.0f; neg[r] = 0.0f;
#pragma unroll
        for (int q = 0; q < TOPK; ++q) t5[r][q] = 0.0f;
    }

    // A fragment base: lane holds rows M=n, K = 4*kc + 2*h + {0,1}
    const float* aBase = As + n * LDS_STRIDE + 2 * h;

    // Two column tiles per K-sweep: one A-fragment load feeds two independent
    // WMMA accumulation chains (halves LDS traffic per wmma, better pipelining).
    for (int t = 0; t < NTILES / WAVES; t += 2) {
        const int jt0 = wave + WAVES * t;
        const int jt1 = wave + WAVES * (t + 1);
        const float* bBase0 = X + (size_t)(jt0 * 16 + n) * D_DIM + 2 * h;
        const float* bBase1 = X + (size_t)(jt1 * 16 + n) * D_DIM + 2 * h;

        v8f acc0 = {0.f,0.f,0.f,0.f,0.f,0.f,0.f,0.f};
        v8f acc1 = {0.f,0.f,0.f,0.f,0.f,0.f,0.f,0.f};
#pragma unroll 4
        for (int kc = 0; kc < KCH; ++kc) {
            v2f a  = *(const v2f*)(aBase  + kc * 4);
            v2f b0 = *(const v2f*)(bBase0 + kc * 4);
            v2f b1 = *(const v2f*)(bBase1 + kc * 4);
            // D = A(16x4,f32) * B(4x16,f32) + C  -> v_wmma_f32_16x16x4_f32
            acc0 = __builtin_amdgcn_wmma_f32_16x16x4_f32(
                false, a, false, b0, (short)0, acc0, false, false);
            acc1 = __builtin_amdgcn_wmma_f32_16x16x4_f32(
                false, a, false, b1, (short)0, acc1, false, false);
        }

        v8f accs[2] = {acc0, acc1};
        int jts[2]  = {jt0, jt1};
#pragma unroll
        for (int p = 0; p < 2; ++p) {
            const bool diagTile = (jts[p] == it);
#pragma unroll
            for (int r = 0; r < 8; ++r) {
                const int m = r + 8 * h;                    // C row in accs[p][r]
                const float e = __expf(accs[p][r] * 10.0f); // exp(sim / 0.1)
                if (n == m) {
                    // same-role column; true diagonal (j==i) masked to 0
                    pos[r] += diagTile ? 0.0f : e;
                } else {
                    neg[r] += e;
                    top5_insert(t5[r], e);
                }
            }
        }
    }

    // dump per-lane partials
#pragma unroll
    for (int r = 0; r < 8; ++r) {
        const int m = r + 8 * h;
        atomicAdd(&negL[m], neg[r]);
        if (n == m) atomicAdd(&posL[m], pos[r]);
        float* tb = &topBuf[(((wave * 16) + m) * 16 + n) * TOPK];
#pragma unroll
        for (int q = 0; q < TOPK; ++q) tb[q] = t5[r][q];
    }
    __syncthreads();

    // merge stage 1: 16 threads per row, each folds 8 (wave,n) 5-lists
    {
        const int row = tid >> 4;
        const int sub = tid & 15;
        float m5[TOPK] = {0.f,0.f,0.f,0.f,0.f};
        for (int p = sub * 8; p < sub * 8 + 8; ++p) {
            const int w = p >> 4, nn = p & 15;
            const float* tb = &topBuf[(((w * 16) + row) * 16 + nn) * TOPK];
#pragma unroll
            for (int q = 0; q < TOPK; ++q) top5_insert(m5, tb[q]);
        }
        float* mb = &mergeBuf[(row * 16 + sub) * TOPK];
#pragma unroll
        for (int q = 0; q < TOPK; ++q) mb[q] = m5[q];
    }
    __syncthreads();

    // merge stage 2 + per-row loss (analytic masks: all rows valid,
    // valid_role_count = 255, valid_count = 4096-255 = 3841)
    if (tid < 16) {
        const int row = tid;
        float m5[TOPK] = {0.f,0.f,0.f,0.f,0.f};
        for (int sub = 0; sub < 16; ++sub) {
            const float* mb = &mergeBuf[(row * 16 + sub) * TOPK];
#pragma unroll
            for (int q = 0; q < TOPK; ++q) top5_insert(m5, mb[q]);
        }
        const float t5sum = m5[0] + m5[1] + m5[2] + m5[3] + m5[4];
        const float p  = posL[row];
        const float ng = negL[row] + t5sum;   // HARD_NEG_WEIGHT=2 -> +1x extra on top5
        const float loss = -logf(p / 255.0f) + logf(ng / 3841.0f);
        atomicAdd(gAcc, loss);
    }
}

__global__ void role_loss_init(float* gAcc) { gAcc[0] = 0.0f; }

__global__ void role_loss_finalize(const float* __restrict__ gAcc,
                                   float* __restrict__ out) {
    out[0] = gAcc[0] / (4096.0f + 1e-10f);
}

extern "C" void kernel_launch(void* const* d_in, const int* in_sizes, int n_in,
                              void* d_out, int out_size, void* d_ws, size_t ws_size,
                              hipStream_t stream) {
    const float* X  = (const float*)d_in[0];   // [256,16,1024] f32, row-major
    float* out      = (float*)d_out;           // scalar loss
    float* gAcc     = (float*)d_ws;            // 1 float accumulator

    role_loss_init<<<1, 1, 0, stream>>>(gAcc);
    role_loss_main<<<NR / 16, 256, 0, stream>>>(X, gAcc);
    role_loss_finalize<<<1, 1, 0, stream>>>(gAcc, out);
}